// SelectiveSSMBackbone_74526272520980
// MI455X (gfx1250) — compile-verified
//
#include <hip/hip_runtime.h>
#include <hip/hip_bf16.h>

// ---------------------------------------------------------------------------
// SelectiveSSMBackbone for MI455X (gfx1250, wave32, WMMA f32 16x16x4)
//   P=32, D=256, NX=16, H_OUT=1, B=8, L=2048  (NT = B*L = 16384 tokens)
// Pipeline:
//   front : h = y@Win+b ; hn = LN(h) ; delta = softplus(..) ; Bm,Cm = hn@W^T
//   scan1 : per (b,d,chunk) intra-chunk state + decay product   (16 chunks)
//   scan2 : serial combine over chunks -> initial state per chunk
//   scan3 : replay chunk with correct init, emit upd[b,l,d]
//   back  : hf = LN(h+upd) ; out = gelu(hf@W1+b1)@W2+b2
//           W1 staged into LDS k-slice-wise via async global->LDS DMA.
// All GEMMs use v_wmma_f32_16x16x4_f32 (fp32, matches reference precision).
// ---------------------------------------------------------------------------

typedef __attribute__((ext_vector_type(2))) float v2f;
typedef __attribute__((ext_vector_type(8))) float v8f;

#define DD   256
#define PP   32
#define NXX  16
#define LL   2048
#define NCH  16
#define LC   (LL / NCH)
#define LDSS 257   // padded row stride (floats) for activation tiles
#define WSTR 264   // W1 slice row stride: 16B-aligned rows + conflict-free B reads

// ---- CDNA5 async global->LDS copy (ASYNCcnt path), with safe fallback ----
#ifndef HAS_ASYNC_B128
#  if defined(__has_builtin)
#    if __has_builtin(__builtin_amdgcn_global_load_async_to_lds_b128)
#      define HAS_ASYNC_B128 1
#    else
#      define HAS_ASYNC_B128 0
#    endif
#  else
#    define HAS_ASYNC_B128 0
#  endif
#endif

// 16-byte payload type exactly as the builtin expects (from clang diagnostic):
//   int __attribute__((vector_size(16)))  in AS1 (src) / AS3 (dst)
typedef int b128i __attribute__((vector_size(16)));
typedef __attribute__((address_space(1))) b128i* gptr_b128;
typedef __attribute__((address_space(3))) b128i* lptr_b128;

__device__ __forceinline__ void async_copy16(const float* gsrc, float* ldst) {
#if HAS_ASYNC_B128
  __builtin_amdgcn_global_load_async_to_lds_b128(
      (gptr_b128)(uintptr_t)gsrc,
      (lptr_b128)(uint32_t)(uintptr_t)ldst,
      /*offset=*/0, /*cpol=*/0);
#else
  *(float4*)ldst = *(const float4*)gsrc;
#endif
}

__device__ __forceinline__ void async_join() {
#if HAS_ASYNC_B128
#  if defined(__has_builtin) && __has_builtin(__builtin_amdgcn_s_wait_asynccnt)
  __builtin_amdgcn_s_wait_asynccnt(0);
#  else
  asm volatile("s_wait_asynccnt 0x0" ::: "memory");
#  endif
#endif
}

__device__ __forceinline__ v8f wmma4(v2f a, v2f b, v8f c) {
  // D = A(16x4,f32) * B(4x16,f32) + C(16x16,f32)
  return __builtin_amdgcn_wmma_f32_16x16x4_f32(
      /*neg_a=*/false, a, /*neg_b=*/false, b,
      /*c_mod=*/(short)0, c, /*reuse_a=*/false, /*reuse_b=*/false);
}

__device__ __forceinline__ v8f zero8() {
  v8f z = {0.f, 0.f, 0.f, 0.f, 0.f, 0.f, 0.f, 0.f};
  return z;
}

__device__ __forceinline__ float wred(float v) {
  v += __shfl_xor(v, 16, 32);
  v += __shfl_xor(v,  8, 32);
  v += __shfl_xor(v,  4, 32);
  v += __shfl_xor(v,  2, 32);
  v += __shfl_xor(v,  1, 32);
  return v;
}

__device__ __forceinline__ float softplus_f(float x) {
  return (x > 20.f) ? x : log1pf(expf(x));
}

// ---------------------------------------------------------------------------
// front: per 16-token tile (one wave):  h, hn, delta, Bm, Cm
// block = 64 threads (2 waves), grid = NT/32
// ---------------------------------------------------------------------------
__global__ __launch_bounds__(64) void ssm_front(
    const float* __restrict__ y,   const float* __restrict__ Win,
    const float* __restrict__ bin, const float* __restrict__ ng,
    const float* __restrict__ nb,  const float* __restrict__ WB,
    const float* __restrict__ WC,  const float* __restrict__ qd,
    const float* __restrict__ pdl,
    float* __restrict__ h, float* __restrict__ hn,
    float* __restrict__ dlt, float* __restrict__ Bm, float* __restrict__ Cm)
{
  __shared__ float tile[2][16][LDSS];
  const int wave = threadIdx.x >> 5;
  const int lane = threadIdx.x & 31;
  const int t16  = lane & 15;
  const int half = lane >> 4;
  const int kb   = half * 2;
  const int tok0 = (blockIdx.x * 2 + wave) * 16;

  // ---- GEMM: h(16x256) = y(16x32) @ Win(32x256) + bin ----
  v8f acc[16];
#pragma unroll
  for (int nt = 0; nt < 16; ++nt) acc[nt] = zero8();
#pragma unroll
  for (int kk = 0; kk < 8; ++kk) {             // K = 32 = 8 * 4
    const int k0 = kk * 4 + kb;
    v2f a;
    a.x = y[(size_t)(tok0 + t16) * PP + k0];
    a.y = y[(size_t)(tok0 + t16) * PP + k0 + 1];
#pragma unroll
    for (int nt = 0; nt < 16; ++nt) {
      v2f b;
      b.x = Win[(size_t)k0 * DD + nt * 16 + t16];
      b.y = Win[(size_t)(k0 + 1) * DD + nt * 16 + t16];
      acc[nt] = wmma4(a, b, acc[nt]);
    }
  }
  // bias + store h to global and LDS (D-layout: m = r + 8*half, n = t16)
#pragma unroll
  for (int nt = 0; nt < 16; ++nt) {
    const int col = nt * 16 + t16;
    const float bv = bin[col];
#pragma unroll
    for (int r = 0; r < 8; ++r) {
      const int m = r + 8 * half;
      const float v = acc[nt][r] + bv;
      h[(size_t)(tok0 + m) * DD + col] = v;
      tile[wave][m][col] = v;
    }
  }
  __syncthreads();

  // ---- per-token layernorm + delta ----
  const float pdelta = pdl[0];
  for (int tk = 0; tk < 16; ++tk) {
    float v[8];
    float s = 0.f, ss = 0.f;
#pragma unroll
    for (int j = 0; j < 8; ++j) {
      v[j] = tile[wave][tk][lane + 32 * j];
      s  += v[j];
      ss += v[j] * v[j];
    }
    s  = wred(s);
    ss = wred(ss);
    const float mu = s * (1.f / 256.f);
    const float rs = rsqrtf(ss * (1.f / 256.f) - mu * mu + 1e-5f);
    float dq = 0.f;
#pragma unroll
    for (int j = 0; j < 8; ++j) {
      const int idx = lane + 32 * j;
      const float hv = (v[j] - mu) * rs * ng[idx] + nb[idx];
      tile[wave][tk][idx] = hv;                       // overwrite with hn
      hn[(size_t)(tok0 + tk) * DD + idx] = hv;
      dq = fmaf(hv, qd[idx], dq);
    }
    dq = wred(dq);
    if (lane == 0) dlt[tok0 + tk] = softplus_f(pdelta + dq);
  }
  __syncthreads();

  // ---- Bm = hn @ W_B^T, Cm = hn @ W_C^T  (16x256 @ 256x16, via WMMA) ----
  v8f accB = zero8(), accC = zero8();
  for (int kk = 0; kk < 64; ++kk) {            // K = 256 = 64 * 4
    const int k0 = kk * 4 + kb;
    v2f a;
    a.x = tile[wave][t16][k0];
    a.y = tile[wave][t16][k0 + 1];
    v2f b, c;
    b.x = WB[(size_t)t16 * DD + k0];  b.y = WB[(size_t)t16 * DD + k0 + 1];
    c.x = WC[(size_t)t16 * DD + k0];  c.y = WC[(size_t)t16 * DD + k0 + 1];
    accB = wmma4(a, b, accB);
    accC = wmma4(a, c, accC);
  }
#pragma unroll
  for (int r = 0; r < 8; ++r) {
    const int m = r + 8 * half;
    Bm[(size_t)(tok0 + m) * NXX + t16] = accB[r];
    Cm[(size_t)(tok0 + m) * NXX + t16] = accC[r];
  }
}

// ---------------------------------------------------------------------------
// scan1: intra-chunk local scan + decay product.
// grid = B*NCH blocks (b,ch uniform per block), block = 256 threads (d)
// ---------------------------------------------------------------------------
__global__ __launch_bounds__(256) void ssm_scan1(
    const float* __restrict__ A,  const float* __restrict__ hn,
    const float* __restrict__ dlt, const float* __restrict__ Bm,
    float* __restrict__ xend, float* __restrict__ Pend)
{
  const int d  = threadIdx.x;
  const int b  = blockIdx.x >> 4;
  const int ch = blockIdx.x & 15;
  float Ar[NXX], x[NXX], Pp[NXX];
#pragma unroll
  for (int n = 0; n < NXX; ++n) {
    Ar[n] = A[(size_t)d * NXX + n];
    x[n] = 0.f;
    Pp[n] = 1.f;
  }
  const int l0 = ch * LC;
  for (int t = 0; t < LC; ++t) {
    const size_t base = (size_t)b * LL + l0 + t;
    const float de = dlt[base];
    const float du = de * hn[base * DD + d];
#pragma unroll
    for (int n = 0; n < NXX; ++n) {
      const float dA = expf(de * Ar[n]);
      Pp[n] *= dA;
      x[n] = fmaf(dA, x[n], du * Bm[base * NXX + n]);
    }
  }
  const size_t o = ((size_t)(b * DD + d) * NCH + ch) * NXX;
#pragma unroll
  for (int n = 0; n < NXX; ++n) {
    xend[o + n] = x[n];
    Pend[o + n] = Pp[n];
  }
}

// ---------------------------------------------------------------------------
// scan2: serial combine over 16 chunks per (b,d,n)  -> initial states
// grid = B*NCH blocks of 256 threads (B*D*NX total threads)
// ---------------------------------------------------------------------------
__global__ __launch_bounds__(256) void ssm_scan2(
    const float* __restrict__ xend, const float* __restrict__ Pend,
    float* __restrict__ xinit)
{
  const int tid = blockIdx.x * 256 + threadIdx.x;
  const int bd = tid >> 4;
  const int n  = tid & 15;
  float xg = 0.f;
  for (int ch = 0; ch < NCH; ++ch) {
    const size_t idx = ((size_t)bd * NCH + ch) * NXX + n;
    xinit[idx] = xg;
    xg = fmaf(Pend[idx], xg, xend[idx]);
  }
}

// ---------------------------------------------------------------------------
// scan3: replay chunk with correct initial state, emit upd = sum_n x*C
// ---------------------------------------------------------------------------
__global__ __launch_bounds__(256) void ssm_scan3(
    const float* __restrict__ A,   const float* __restrict__ hn,
    const float* __restrict__ dlt, const float* __restrict__ Bm,
    const float* __restrict__ Cm,  const float* __restrict__ xinit,
    float* __restrict__ upd)
{
  const int d  = threadIdx.x;
  const int b  = blockIdx.x >> 4;
  const int ch = blockIdx.x & 15;
  float Ar[NXX], x[NXX];
  const size_t o = ((size_t)(b * DD + d) * NCH + ch) * NXX;
#pragma unroll
  for (int n = 0; n < NXX; ++n) {
    Ar[n] = A[(size_t)d * NXX + n];
    x[n] = xinit[o + n];
  }
  const int l0 = ch * LC;
  for (int t = 0; t < LC; ++t) {
    const size_t base = (size_t)b * LL + l0 + t;
    const float de = dlt[base];
    const float du = de * hn[base * DD + d];
    float yv = 0.f;
#pragma unroll
    for (int n = 0; n < NXX; ++n) {
      const float dA = expf(de * Ar[n]);
      x[n] = fmaf(dA, x[n], du * Bm[base * NXX + n]);
      yv = fmaf(x[n], Cm[base * NXX + n], yv);
    }
    upd[base * DD + d] = yv;
  }
}

// ---------------------------------------------------------------------------
// back: hf = LN(h+upd); out = gelu(hf@W1+b1)@W2+b2
// block = 64 threads (2 waves), grid = NT/32, one wave per 16-token tile.
// W1 is staged k-slice-wise (16 rows = 16KB) into LDS with async DMA and
// shared by both waves; B-fragments for GEMM1 then come from LDS.
// ---------------------------------------------------------------------------
__global__ __launch_bounds__(64) void ssm_back(
    const float* __restrict__ h,   const float* __restrict__ upd,
    const float* __restrict__ nfg, const float* __restrict__ nfb,
    const float* __restrict__ W1,  const float* __restrict__ b1,
    const float* __restrict__ W2,  const float* __restrict__ b2,
    float* __restrict__ out)
{
  __shared__ float tile[2][16][LDSS];
  __shared__ __align__(16) float ldsW[16][WSTR];   // one 16-row W1 k-slice
  const int wave = threadIdx.x >> 5;
  const int lane = threadIdx.x & 31;
  const int t16  = lane & 15;
  const int half = lane >> 4;
  const int kb   = half * 2;
  const int tok0 = (blockIdx.x * 2 + wave) * 16;

  // residual + layernorm into LDS
  for (int tk = 0; tk < 16; ++tk) {
    float v[8];
    float s = 0.f, ss = 0.f;
#pragma unroll
    for (int j = 0; j < 8; ++j) {
      const int idx = lane + 32 * j;
      const size_t g = (size_t)(tok0 + tk) * DD + idx;
      v[j] = h[g] + upd[g];
      s  += v[j];
      ss += v[j] * v[j];
    }
    s  = wred(s);
    ss = wred(ss);
    const float mu = s * (1.f / 256.f);
    const float rs = rsqrtf(ss * (1.f / 256.f) - mu * mu + 1e-5f);
#pragma unroll
    for (int j = 0; j < 8; ++j) {
      const int idx = lane + 32 * j;
      tile[wave][tk][idx] = (v[j] - mu) * rs * nfg[idx] + nfb[idx];
    }
  }
  __syncthreads();

  // ---- GEMM1: g(16x256) = hf(16x256) @ W1(256x256), W1 via LDS slices ----
  v8f acc[16];
#pragma unroll
  for (int nt = 0; nt < 16; ++nt) acc[nt] = zero8();

  for (int s = 0; s < 16; ++s) {               // 16 k-slices of 16 rows
    // stage W1 rows [16s, 16s+16): 1024 16B-chunks, 64 threads -> 16 each
    const float* gW = W1 + (size_t)s * 16 * DD;
#pragma unroll
    for (int c = 0; c < 16; ++c) {
      const int chunk = threadIdx.x + 64 * c;  // 0..1023
      const int row = chunk >> 6;              // 0..15
      const int cc  = chunk & 63;              // 16B chunk within row
      async_copy16(gW + (size_t)row * DD + cc * 4, &ldsW[row][cc * 4]);
    }
    async_join();
    __syncthreads();

#pragma unroll
    for (int kk2 = 0; kk2 < 4; ++kk2) {        // 4 wmma k-steps per slice
      const int kl = kk2 * 4 + kb;             // row within slice
      const int k0 = s * 16 + kl;              // absolute k
      v2f a;
      a.x = tile[wave][t16][k0];
      a.y = tile[wave][t16][k0 + 1];
#pragma unroll
      for (int nt = 0; nt < 16; ++nt) {
        v2f b;
        b.x = ldsW[kl][nt * 16 + t16];
        b.y = ldsW[kl + 1][nt * 16 + t16];
        acc[nt] = wmma4(a, b, acc[nt]);
      }
    }
    __syncthreads();                            // before overwriting slice
  }

  // bias + exact gelu, write g back to LDS (A-layout staging)
#pragma unroll
  for (int nt = 0; nt < 16; ++nt) {
    const int col = nt * 16 + t16;
    const float bb = b1[col];
#pragma unroll
    for (int r = 0; r < 8; ++r) {
      const float xv = acc[nt][r] + bb;
      const float gv = 0.5f * xv * (1.f + erff(xv * 0.70710678118654752f));
      tile[wave][r + 8 * half][col] = gv;
    }
  }
  __syncthreads();

  // ---- GEMM2: out(16x32) = g(16x256) @ W2(256x32) + b2 ----
  v8f acc2[2];
  acc2[0] = zero8();
  acc2[1] = zero8();
  for (int kk = 0; kk < 64; ++kk) {
    const int k0 = kk * 4 + kb;
    v2f a;
    a.x = tile[wave][t16][k0];
    a.y = tile[wave][t16][k0 + 1];
#pragma unroll
    for (int nt = 0; nt < 2; ++nt) {
      v2f b;
      b.x = W2[(size_t)k0 * PP + nt * 16 + t16];
      b.y = W2[(size_t)(k0 + 1) * PP + nt * 16 + t16];
      acc2[nt] = wmma4(a, b, acc2[nt]);
    }
  }
#pragma unroll
  for (int nt = 0; nt < 2; ++nt) {
    const int col = nt * 16 + t16;
    const float bb = b2[col];
#pragma unroll
    for (int r = 0; r < 8; ++r) {
      out[(size_t)(tok0 + r + 8 * half) * PP + col] = acc2[nt][r] + bb;
    }
  }
}

// ---------------------------------------------------------------------------
extern "C" void kernel_launch(void* const* d_in, const int* in_sizes, int n_in,
                              void* d_out, int out_size, void* d_ws, size_t ws_size,
                              hipStream_t stream) {
  (void)n_in; (void)out_size; (void)ws_size;
  const float* y    = (const float*)d_in[0];
  const float* Win  = (const float*)d_in[1];
  const float* bin_ = (const float*)d_in[2];
  const float* ng   = (const float*)d_in[3];
  const float* nb   = (const float*)d_in[4];
  const float* A    = (const float*)d_in[5];
  const float* W_B  = (const float*)d_in[6];
  const float* W_C  = (const float*)d_in[7];
  const float* qd   = (const float*)d_in[8];
  const float* pdl  = (const float*)d_in[9];
  const float* nfg  = (const float*)d_in[10];
  const float* nfb  = (const float*)d_in[11];
  const float* W1   = (const float*)d_in[12];
  const float* b1   = (const float*)d_in[13];
  const float* W2   = (const float*)d_in[14];
  const float* b2   = (const float*)d_in[15];
  float* out = (float*)d_out;

  const int NT = in_sizes[0] / PP;   // tokens = B*L
  const int Bb = NT / LL;            // batch

  float* ws = (float*)d_ws;
  size_t o = 0;
  float* h    = ws + o;  o += (size_t)NT * DD;
  float* hn   = ws + o;  o += (size_t)NT * DD;
  float* dlt  = ws + o;  o += (size_t)NT;
  float* Bm   = ws + o;  o += (size_t)NT * NXX;
  float* Cm   = ws + o;  o += (size_t)NT * NXX;
  float* upd  = ws + o;  o += (size_t)NT * DD;
  const size_t scansz = (size_t)Bb * DD * NCH * NXX;
  float* xend  = ws + o; o += scansz;
  float* Pend  = ws + o; o += scansz;
  float* xinit = ws + o; o += scansz;

  ssm_front<<<NT / 32, 64, 0, stream>>>(y, Win, bin_, ng, nb, W_B, W_C, qd, pdl,
                                        h, hn, dlt, Bm, Cm);
  ssm_scan1<<<Bb * NCH, 256, 0, stream>>>(A, hn, dlt, Bm, xend, Pend);
  ssm_scan2<<<Bb * NCH, 256, 0, stream>>>(xend, Pend, xinit);
  ssm_scan3<<<Bb * NCH, 256, 0, stream>>>(A, hn, dlt, Bm, Cm, xinit, upd);
  ssm_back<<<NT / 32, 64, 0, stream>>>(h, upd, nfg, nfb, W1, b1, W2, b2, out);
}